// TensorDLT_73031623901550
// MI455X (gfx1250) — compile-verified
//
#include <hip/hip_runtime.h>
#include <stdint.h>

// ---------------------------------------------------------------------------
// TensorDLT for MI455X (gfx1250).
//
// Math shortcut: M = An^T An is rank<=8 PSD (An is 8x9), so eigh(M)[...,0]
// is exactly the null vector of An; the final H/H[2,2] normalization removes
// scale and sign. We compute it with unrolled Householder QR + backsolve.
//
// Data movement: per-wave TDM (tensor_load_to_lds) stages 32 batches x 72 f32
// of noise (9 KB) into LDS, overlapped with point normalization; completion
// via s_wait_tensorcnt; consumed with ds_load_b128. Wave-local -> no barrier.
//
// Fast math: v_sqrt_f32 / v_rcp_f32 (~1 ulp) instead of IEEE expansions —
// the reference adds 1e-4 noise, so 1e-7 relative error is far below the
// problem's own perturbation floor.
// ---------------------------------------------------------------------------

typedef __attribute__((ext_vector_type(4))) unsigned int tdm_u32x4;
typedef __attribute__((ext_vector_type(8))) int          tdm_i32x8;
typedef __attribute__((ext_vector_type(4))) int          tdm_i32x4;

#define TPB  128
#define NPB  72     // noise floats per batch
#define WAVE 32

__device__ __forceinline__ float frcp(float x)  { return __builtin_amdgcn_rcpf(x); }
__device__ __forceinline__ float fsqrt(float x) { return __builtin_amdgcn_sqrtf(x); }

__device__ __forceinline__ void hartley(const float4 p01, const float4 p23,
                                        float X[4], float Y[4],
                                        float& s, float& d, float& cx, float& cy)
{
    float x0 = p01.x, y0 = p01.y, x1 = p01.z, y1 = p01.w;
    float x2 = p23.x, y2 = p23.y, x3 = p23.z, y3 = p23.w;
    cx = 0.25f * (x0 + x1 + x2 + x3);
    cy = 0.25f * (y0 + y1 + y2 + y3);
    float dx0 = x0 - cx, dy0 = y0 - cy, dx1 = x1 - cx, dy1 = y1 - cy;
    float dx2 = x2 - cx, dy2 = y2 - cy, dx3 = x3 - cx, dy3 = y3 - cy;
    d = 0.25f * (fsqrt(dx0 * dx0 + dy0 * dy0) + fsqrt(dx1 * dx1 + dy1 * dy1) +
                 fsqrt(dx2 * dx2 + dy2 * dy2) + fsqrt(dx3 * dx3 + dy3 * dy3));
    s = 1.41421356237309515f * frcp(d + 1e-8f);
    X[0] = dx0 * s; Y[0] = dy0 * s;
    X[1] = dx1 * s; Y[1] = dy1 * s;
    X[2] = dx2 * s; Y[2] = dy2 * s;
    X[3] = dx3 * s; Y[3] = dy3 * s;
}

__global__ void __launch_bounds__(TPB)
tensordlt_kernel(const float* __restrict__ xA,
                 const float* __restrict__ xB,
                 const float* __restrict__ noise,
                 float* __restrict__ out,
                 int nbatch)
{
    __shared__ float lnoise[TPB * NPB];   // 36 KB / workgroup

    const int tid = (int)threadIdx.x;
    const int b   = (int)blockIdx.x * TPB + tid;

    // ---- 1) Per-wave async TDM stage of this wave's noise slice ----------
    {
        const int wv = __builtin_amdgcn_readfirstlane(tid >> 5);     // scalar
        const uint32_t ne = WAVE * NPB;                               // 2304 f32
        const uint64_t ga = (uint64_t)(uintptr_t)
            (noise + ((uint64_t)blockIdx.x * TPB + (uint64_t)wv * WAVE) * NPB);
        const uint32_t la = (uint32_t)(uintptr_t)(&lnoise[0]) + (uint32_t)wv * ne * 4u;

        tdm_u32x4 g0;
        g0[0] = 1u;                                             // D# count=1, user mode
        g0[1] = la;                                             // lds_addr (bytes)
        g0[2] = (uint32_t)ga;                                   // global_addr[31:0]
        g0[3] = ((uint32_t)(ga >> 32) & 0x01FFFFFFu)            // global_addr[56:32]
              | (2u << 30);                                     // type=2
        tdm_i32x8 g1;
        g1[0] = (int)(2u << 16);                                // data_size = 4 B
        g1[1] = (int)((ne & 0xFFFFu) << 16);                    // tensor_dim0[15:0]
        g1[2] = (int)(((ne >> 16) & 0xFFFFu) | (1u << 16));     // dim0 hi | tensor_dim1=1
        g1[3] = (int)((ne & 0xFFFFu) << 16);                    // tile_dim0 = ne
        g1[4] = 0;                                              // tile_dim1/2 unused
        g1[5] = (int)ne;                                        // tensor_dim0_stride
        g1[6] = 0;
        g1[7] = 0;
        tdm_i32x4 gz4 = {0, 0, 0, 0};
        tdm_i32x8 gz8 = {0, 0, 0, 0, 0, 0, 0, 0};
        __builtin_amdgcn_tensor_load_to_lds(g0, g1, gz4, gz4, gz8, 0);
    }

    if (b >= nbatch) { __builtin_amdgcn_s_wait_tensorcnt(0); return; }

    // ---- 2) Overlap with DMA: load correspondences, normalize ------------
    const float4* pA = reinterpret_cast<const float4*>(xA) + (size_t)b * 2;
    const float4* pB = reinterpret_cast<const float4*>(xB) + (size_t)b * 2;
    float4 a01 = pA[0], a23 = pA[1];
    float4 b01 = pB[0], b23 = pB[1];

    float X1[4], Y1[4], X2[4], Y2[4];
    float s1, d1, cx1, cy1, s2, d2, cx2, cy2;
    hartley(a01, a23, X1, Y1, s1, d1, cx1, cy1);
    hartley(b01, b23, X2, Y2, s2, d2, cx2, cy2);

    // ---- 3) Build the 8x9 DLT matrix ------------------------------------
    float A[8][9];
#pragma unroll
    for (int i = 0; i < 4; ++i) {
        float u1 = X1[i], v1 = Y1[i], u2 = X2[i], v2 = Y2[i];
        A[2 * i + 0][0] = 0.f;      A[2 * i + 0][1] = 0.f;      A[2 * i + 0][2] = 0.f;
        A[2 * i + 0][3] = -u1;      A[2 * i + 0][4] = -v1;      A[2 * i + 0][5] = -1.f;
        A[2 * i + 0][6] = v2 * u1;  A[2 * i + 0][7] = v2 * v1;  A[2 * i + 0][8] = v2;
        A[2 * i + 1][0] = u1;       A[2 * i + 1][1] = v1;       A[2 * i + 1][2] = 1.f;
        A[2 * i + 1][3] = 0.f;      A[2 * i + 1][4] = 0.f;      A[2 * i + 1][5] = 0.f;
        A[2 * i + 1][6] = -u2 * u1; A[2 * i + 1][7] = -u2 * v1; A[2 * i + 1][8] = -u2;
    }

    // ---- 4) DMA complete -> add 1e-4 * noise from LDS --------------------
    __builtin_amdgcn_s_wait_tensorcnt(0);
    const float4* np = reinterpret_cast<const float4*>(&lnoise[(unsigned)tid * NPB]);
    float* Af = &A[0][0];
#pragma unroll
    for (int q = 0; q < 18; ++q) {
        float4 nv = np[q];
        Af[4 * q + 0] = fmaf(1e-4f, nv.x, Af[4 * q + 0]);
        Af[4 * q + 1] = fmaf(1e-4f, nv.y, Af[4 * q + 1]);
        Af[4 * q + 2] = fmaf(1e-4f, nv.z, Af[4 * q + 2]);
        Af[4 * q + 3] = fmaf(1e-4f, nv.w, Af[4 * q + 3]);
    }

    // ---- 5) Householder QR (8x9), fully unrolled, no pivoting ------------
#pragma unroll
    for (int k = 0; k < 8; ++k) {
        float sig = 0.f;
#pragma unroll
        for (int i = k; i < 8; ++i) sig += A[i][k] * A[i][k];
        float nrm   = fsqrt(sig);
        float akk   = A[k][k];
        float alpha = (akk >= 0.f) ? -nrm : nrm;
        float vk    = akk - alpha;
        float vtv   = sig - akk * akk + vk * vk + 1e-30f;
        float tau   = 2.0f * frcp(vtv);
#pragma unroll
        for (int j = k + 1; j < 9; ++j) {
            float s = vk * A[k][j];
#pragma unroll
            for (int i = k + 1; i < 8; ++i) s += A[i][k] * A[i][j];
            s *= tau;
            A[k][j] -= s * vk;
#pragma unroll
            for (int i = k + 1; i < 8; ++i) A[i][j] -= s * A[i][k];
        }
        A[k][k] = alpha;   // R diagonal; sub-diagonal holds v (unused later)
    }

    // ---- 6) Back-substitute R h = 0 with h[8] = 1 ------------------------
    float h[9];
    h[8] = 1.0f;
#pragma unroll
    for (int k = 7; k >= 0; --k) {
        float s = 0.f;
#pragma unroll
        for (int j = k + 1; j < 9; ++j) s += A[k][j] * h[j];
        h[k] = -s * frcp(A[k][k]);
    }

    // ---- 7) Un-normalize: F = inv(T_B) * Hn * T_A, scale by F[2][2] ------
    float G[3][3];
#pragma unroll
    for (int r = 0; r < 3; ++r) {
        float g0 = s1 * h[3 * r + 0];
        float g1 = s1 * h[3 * r + 1];
        float g2 = h[3 * r + 2] - cx1 * g0 - cy1 * g1;
        G[r][0] = g0; G[r][1] = g1; G[r][2] = g2;
    }
    // 1/s2 computed algebraically: s2 = sqrt(2)/(d2+1e-8)  =>  1/s2 = (d2+1e-8)/sqrt(2)
    float rs2 = (d2 + 1e-8f) * 0.70710678118654752f;
    float F[3][3];
#pragma unroll
    for (int c = 0; c < 3; ++c) {
        F[0][c] = rs2 * G[0][c] + cx2 * G[2][c];
        F[1][c] = rs2 * G[1][c] + cy2 * G[2][c];
        F[2][c] = G[2][c];
    }
    float w = frcp(F[2][2] + 1e-8f);

    float* o = out + (size_t)b * 9;
#pragma unroll
    for (int r = 0; r < 3; ++r)
#pragma unroll
        for (int c = 0; c < 3; ++c)
            o[3 * r + c] = F[r][c] * w;
}

extern "C" void kernel_launch(void* const* d_in, const int* in_sizes, int n_in,
                              void* d_out, int out_size, void* d_ws, size_t ws_size,
                              hipStream_t stream)
{
    const float* xA    = (const float*)d_in[0];
    const float* xB    = (const float*)d_in[1];
    const float* noise = (const float*)d_in[2];
    float* out         = (float*)d_out;
    const int nbatch   = in_sizes[0] / 8;            // B*4*2 floats
    const int grid     = (nbatch + TPB - 1) / TPB;
    tensordlt_kernel<<<grid, TPB, 0, stream>>>(xA, xB, noise, out, nbatch);
}